// SOM_16484084483472
// MI455X (gfx1250) — compile-verified
//
#include <hip/hip_runtime.h>
#include <hip/hip_bf16.h>

typedef float v2f __attribute__((ext_vector_type(2)));
typedef float v4f __attribute__((ext_vector_type(4)));
typedef float v8f __attribute__((ext_vector_type(8)));

#define M_GRID   128
#define N_GRID   128
#define DIM      2048
#define NCELLS   (M_GRID * N_GRID)       // 16384
#define NITER_F  1000.0f
#define ALPHA_C  0.3f
#define SIGMA_C  64.0f
#define PD_EPS   1e-6f

// ---------------------------------------------------------------------------
// Kernel 0: initialize packed argmin key (dist_bits<<32 | row) to max.
// ---------------------------------------------------------------------------
__global__ void som_init_key(unsigned long long* key) {
    if (threadIdx.x == 0) *key = 0xFFFFFFFFFFFFFFFFull;
}

// ---------------------------------------------------------------------------
// Kernel 1: BMU search. Each wave owns 16 rows; per K-chunk of 4 the wave
// builds the diff tile (x - w + eps) in the f32 16x4 A layout and issues
// v_wmma_f32_16x16x4_f32 with A==B, so the accumulator diagonal collects
// per-row squared distances. x is staged in LDS once per block.
//
// A-tile layout (ISA 7.12.2, 32-bit A 16x4): lane L<16 -> row L, K={0,1};
// lane L+16 -> row L, K={2,3}. The 4x16 B layout for B = A^T is identical,
// so the same VGPR pair feeds both operands.
// C/D diagonal (i,i): i<8 -> lane i, VGPR i;  i>=8 -> lane i+16, VGPR i-8.
// ---------------------------------------------------------------------------
__global__ void __launch_bounds__(256) som_bmu(
        const float* __restrict__ x,
        const float* __restrict__ w,
        unsigned long long* __restrict__ key) {
    __shared__ float xs[DIM];
    {   // cooperative coalesced stage of x into LDS (256 thr * 2 * b128)
        const v4f* x4 = (const v4f*)x;
        v4f* xs4 = (v4f*)xs;
        for (int t = threadIdx.x; t < DIM / 4; t += 256) xs4[t] = x4[t];
    }
    __syncthreads();

    const int lane    = threadIdx.x & 31;
    const int wave    = threadIdx.x >> 5;
    const int rowBase = blockIdx.x * 128 + wave * 16;   // 8 waves * 16 rows
    const int row     = rowBase + (lane & 15);
    const int koff    = (lane >> 4) * 2;                // K sub-offset 0 or 2

    const v2f* wrow = (const v2f*)(w + (size_t)row * DIM);

    v8f c = {0.f, 0.f, 0.f, 0.f, 0.f, 0.f, 0.f, 0.f};
    #pragma unroll 8
    for (int k = 0; k < DIM; k += 4) {
        const int kk = k + koff;
        v2f wv = wrow[kk >> 1];                          // global_load_b64
        v2f xv = *(const v2f*)&xs[kk];                   // ds_load_b64
        v2f a;
        a.x = (xv.x - wv.x) + PD_EPS;                    // PairwiseDistance eps
        a.y = (xv.y - wv.y) + PD_EPS;
        // D = A * A^T + C : diagonal accumulates sum of squared diffs
        c = __builtin_amdgcn_wmma_f32_16x16x4_f32(
                false, a, false, a, (short)0, c, false, false);
    }

    // Extract diagonal: lanes 0-7 hold rows 0-7 in c[lane],
    // lanes 24-31 hold rows 8-15 in c[lane-24].
    const bool diagLane = (lane < 8) || (lane >= 24);
    const int  s  = lane & 7;                            // VGPR index
    const float d2 = s==0?c[0]:s==1?c[1]:s==2?c[2]:s==3?c[3]
                   : s==4?c[4]:s==5?c[5]:s==6?c[6]:c[7];
    if (diagLane) {
        const int r = rowBase + ((lane < 8) ? lane : (lane - 16));
        // d2 >= 0 so its IEEE bits order as unsigned; low index wins ties
        // (== first-min semantics of argmin). sqrt is monotone -> same argmin.
        const unsigned long long packed =
            ((unsigned long long)__float_as_uint(d2) << 32) | (unsigned)r;
        atomicMin(key, packed);                          // global_atomic_min_u64
    }
}

// ---------------------------------------------------------------------------
// Kernel 2: streaming update. One row per block; rate is uniform per row.
// W loads stay regular-temporal (pass 1 left W resident in the 192MB L2);
// output uses nontemporal stores (never re-read).
// ---------------------------------------------------------------------------
__global__ void __launch_bounds__(256) som_update(
        const float* __restrict__ x,
        const float* __restrict__ w,
        const int*   __restrict__ itp,
        const unsigned long long* __restrict__ key,
        float* __restrict__ out) {
    const int row = blockIdx.x;

    const float lr       = 1.0f - (float)(*itp) / NITER_F;
    const float alpha_op = ALPHA_C * lr;
    const float sigma_op = SIGMA_C * lr;

    const unsigned bmu = (unsigned)(*key & 0xFFFFFFFFull);
    const int bi = (int)(bmu >> 7), bj = (int)(bmu & 127);
    const int i  = row >> 7,        j  = row & 127;
    const float di = (float)(i - bi), dj = (float)(j - bj);
    const float rate =
        alpha_op * __expf(-(di * di + dj * dj) / (sigma_op * sigma_op));

    const v4f* x4 = (const v4f*)x;
    const v4f* w4 = (const v4f*)(w + (size_t)row * DIM);
    v4f*       o4 = (v4f*)(out + (size_t)row * DIM);

    #pragma unroll
    for (int t = threadIdx.x; t < DIM / 4; t += 256) {   // 2 iters
        const v4f wv = w4[t];                             // global_load_b128
        const v4f xv = x4[t];
        v4f ov;
        ov.x = fmaf(rate, xv.x - wv.x, wv.x);
        ov.y = fmaf(rate, xv.y - wv.y, wv.y);
        ov.z = fmaf(rate, xv.z - wv.z, wv.z);
        ov.w = fmaf(rate, xv.w - wv.w, wv.w);
        __builtin_nontemporal_store(ov, &o4[t]);          // global_store_b128 nt
    }
}

// ---------------------------------------------------------------------------
extern "C" void kernel_launch(void* const* d_in, const int* in_sizes, int n_in,
                              void* d_out, int out_size, void* d_ws, size_t ws_size,
                              hipStream_t stream) {
    const float* x  = (const float*)d_in[0];             // (2048,) f32
    const float* w  = (const float*)d_in[1];             // (16384, 2048) f32
    // d_in[2]: locations (int64 meshgrid) -- recomputed analytically on device
    const int*   it = (const int*)d_in[3];               // scalar int
    float* out = (float*)d_out;
    unsigned long long* key = (unsigned long long*)d_ws;

    som_init_key<<<1, 1, 0, stream>>>(key);
    som_bmu<<<NCELLS / 128, 256, 0, stream>>>(x, w, key);
    som_update<<<NCELLS, 256, 0, stream>>>(x, w, it, key, out);
}